// VGGSmall_new_77034533421674
// MI455X (gfx1250) — compile-verified
//
#include <hip/hip_runtime.h>
#include <stdint.h>

// ---------------------------------------------------------------------------
// AdderNet VGG-Small inference for MI455X (gfx1250, wave32).
//  - 8x fused adder2d(3x3,pad1) + BN(eval) + ReLU (+2x2 maxpool) kernels.
//    L1 distance is not expressible as a matmul, so these ride the VALU
//    (v_sub + v_add with |src| modifier) with full LDS staging for reuse.
//  - Final FC [16,2048]x[2048,10]: async-DMA both operands into LDS
//    (global_load_async_to_lds_b128, ASYNCcnt), then a branch-free
//    V_WMMA_F32_16X16X4_F32 K-loop on one full-EXEC wave32.
// ---------------------------------------------------------------------------

typedef float v2f __attribute__((ext_vector_type(2)));
typedef float v4f __attribute__((ext_vector_type(4)));
typedef float v8f __attribute__((ext_vector_type(8)));

#define BN_EPS 1e-5f

// Async global->LDS copy of 16 bytes (CDNA5 GLOBAL_LOAD_ASYNC_TO_LDS_B128).
// VDST VGPR carries the LDS byte address (generic shared ptr low 32 bits per
// ISA aperture translation); VADDR pair carries the 64-bit global address.
__device__ __forceinline__ void async_g2l_b128(const float* g, const float* l) {
  asm volatile("global_load_async_to_lds_b128 %0, %1, off"
               :: "v"((unsigned)(uintptr_t)l), "v"(g)
               : "memory");
}

__device__ __forceinline__ void wait_async0() {
  asm volatile("s_wait_asynccnt 0x0" ::: "memory");
}

// ---------------------------------------------------------------------------
// Fused adder2d + BN + ReLU (+optional 2x2 maxpool).
// Grid: B*Cout blocks; block = 256 threads; one (b, co) pair per block.
// Dynamic LDS: [ padded input Cin*(H+2)*(W+2) | weights Cin*9 |
//               (POOL only) conv map H*W ]
// Zero padding means out-of-image taps contribute |0 - w| = |w|; staging a
// zero-padded input copy keeps the inner loop branch-free.
// ---------------------------------------------------------------------------
template <bool POOL>
__global__ void adder_bn_relu_kernel(const float* __restrict__ x,
                                     const float* __restrict__ w,
                                     const float* __restrict__ gamma,
                                     const float* __restrict__ beta,
                                     const float* __restrict__ mean,
                                     const float* __restrict__ var,
                                     float* __restrict__ out,
                                     int Cin, int Cout, int H, int W) {
  extern __shared__ float lds[];
  const int b  = blockIdx.x / Cout;
  const int co = blockIdx.x % Cout;
  const int Hp = H + 2, Wp = W + 2;

  float* s_x = lds;                      // Cin * Hp * Wp (zero padded)
  float* s_w = s_x + Cin * Hp * Wp;      // Cin * 9
  float* s_o = s_w + Cin * 9;            // H * W (POOL only)

  // Stage this output channel's filter.
  const float* wco = w + (size_t)co * Cin * 9;
  for (int i = threadIdx.x; i < Cin * 9; i += blockDim.x) s_w[i] = wco[i];

  // Stage zero-padded input image for batch b.
  const float* xb = x + (size_t)b * Cin * H * W;
  const int npad = Cin * Hp * Wp;
  for (int i = threadIdx.x; i < npad; i += blockDim.x) {
    __builtin_prefetch(xb + i, 0, 3);  // global_prefetch_b8 into near caches
    int ci = i / (Hp * Wp);
    int r  = i % (Hp * Wp);
    int ih = r / Wp - 1;
    int iw = r % Wp - 1;
    float v = 0.0f;
    if ((unsigned)ih < (unsigned)H && (unsigned)iw < (unsigned)W)
      v = xb[(ci * H + ih) * W + iw];
    s_x[i] = v;
  }
  __syncthreads();

  // BN(eval) folded constants: y = (-acc) * inv + bias
  const float inv  = gamma[co] / sqrtf(var[co] + BN_EPS);
  const float bias = beta[co] - mean[co] * inv;

  for (int p = threadIdx.x; p < H * W; p += blockDim.x) {
    const int oh = p / W, ow = p % W;
    float acc = 0.0f;
    for (int ci = 0; ci < Cin; ++ci) {
      const float* xr = s_x + (ci * Hp + oh) * Wp + ow;
      const float* wr = s_w + ci * 9;
#pragma unroll
      for (int kh = 0; kh < 3; ++kh) {
#pragma unroll
        for (int kw = 0; kw < 3; ++kw) {
          acc += fabsf(xr[kh * Wp + kw] - wr[kh * 3 + kw]);  // v_sub + v_add |m|
        }
      }
    }
    float y = fmaf(-acc, inv, bias);
    y = fmaxf(y, 0.0f);
    if (POOL) {
      s_o[p] = y;
    } else {
      out[((size_t)(b * Cout + co) * H + oh) * W + ow] = y;
    }
  }

  if (POOL) {
    __syncthreads();
    const int Ho = H >> 1, Wo = W >> 1;
    for (int p = threadIdx.x; p < Ho * Wo; p += blockDim.x) {
      const int oh = p / Wo, ow = p % Wo;
      const float* r0 = s_o + (2 * oh) * W + 2 * ow;
      const float* r1 = r0 + W;
      float m = fmaxf(fmaxf(r0[0], r0[1]), fmaxf(r1[0], r1[1]));
      out[((size_t)(b * Cout + co) * Ho + oh) * Wo + ow] = m;
    }
  }
}

// ---------------------------------------------------------------------------
// FC: out[16,10] = h[16,2048] @ fc_w[10,2048]^T + fc_b.
// Block = 256 threads (8 waves): all waves async-DMA A and padded-B into LDS
// (256 KB, within the 320 KB/WGP CDNA5 budget), then wave 0 (EXEC all ones,
// as WMMA requires) runs the 512-step V_WMMA_F32_16X16X4_F32 K-loop straight
// out of LDS -- no per-iteration branches, no EXEC save/restore.
//
// ISA operand layouts (16x4 f32 A / 4x16 B / 16x16 f32 C-D, wave32):
//   A: lane m in [0,15]  -> v0 = A[m][0], v1 = A[m][1]
//      lane m in [16,31] -> v0 = A[m-16][2], v1 = A[m-16][3]
//   B (mirrored):  lane n in [0,15]  -> v0 = B[0][n], v1 = B[1][n]
//                  lane n in [16,31] -> v0 = B[2][n-16], v1 = B[3][n-16]
//   C/D: VGPR j, lanes 0-15 -> row j, lanes 16-31 -> row j+8; col = lane&15.
// ---------------------------------------------------------------------------
__global__ void fc_wmma_kernel(const float* __restrict__ h,
                               const float* __restrict__ fw,
                               const float* __restrict__ fb,
                               float* __restrict__ out) {
  extern __shared__ float lds[];
  float* s_a = lds;             // 16 * 2048  (activations)
  float* s_b = lds + 16 * 2048; // 16 * 2048  (fc_w rows, rows 10..15 zeroed)
  const int tid = threadIdx.x;  // 0..255

  // Async-DMA A (32768 floats) and the 10 real B rows (20480 floats) to LDS.
  for (int i = tid * 4; i < 16 * 2048; i += 256 * 4)
    async_g2l_b128(h + i, s_a + i);
  for (int i = tid * 4; i < 10 * 2048; i += 256 * 4)
    async_g2l_b128(fw + i, s_b + i);

  // Zero-fill the padded B rows (units 10..15).
  const v4f zero4 = {0.0f, 0.0f, 0.0f, 0.0f};
  for (int i = 10 * 2048 + tid * 4; i < 16 * 2048; i += 256 * 4)
    *(v4f*)(s_b + i) = zero4;

  wait_async0();     // drain ASYNCcnt before any wave touches the LDS tiles
  __syncthreads();

  if (tid < 32) {    // wave 0 only; all 32 lanes active -> EXEC == ~0
    const int lane  = tid;
    const int mn    = lane & 15;   // A row (batch) / B col (output unit)
    const int khalf = lane >> 4;   // K pair {0,1} or {2,3}

    const float* pa = s_a + mn * 2048 + khalf * 2;
    const float* pb = s_b + mn * 2048 + khalf * 2;

    v8f acc = {};
    for (int k = 0; k < 2048; k += 16) {
#pragma unroll
      for (int u = 0; u < 4; ++u) {
        v2f a = *(const v2f*)(pa + k + u * 4);   // ds_load_b64
        v2f b = *(const v2f*)(pb + k + u * 4);   // ds_load_b64
        acc = __builtin_amdgcn_wmma_f32_16x16x4_f32(
            /*neg_a=*/false, a, /*neg_b=*/false, b,
            /*c_mod=*/(short)0, acc, /*reuse_a=*/false, /*reuse_b=*/false);
      }
    }

    // Scatter D + bias: row = j + 8*khalf, col = mn (cols >= 10 are padding).
    if (mn < 10) {
      const float bv = fb[mn];
#pragma unroll
      for (int j = 0; j < 8; ++j) {
        const int row = j + khalf * 8;
        out[row * 10 + mn] = acc[j] + bv;
      }
    }
  }
}

// ---------------------------------------------------------------------------
// Host-side orchestration.
// Input order: x, adder_ws[0..7], bn_gamma[0..7], bn_beta[0..7],
//              bn_mean[0..7], bn_var[0..7], fc_w, fc_b
// ---------------------------------------------------------------------------
struct LayerCfg { int Cin, Cout, H, W, pool; };

static const LayerCfg kCfg[8] = {
    {  3,  32, 32, 32, 0},
    { 32,  32, 32, 32, 1},
    { 32,  64, 16, 16, 0},
    { 64,  64, 16, 16, 1},
    { 64, 128,  8,  8, 0},
    {128, 128,  8,  8, 0},
    {128, 128,  8,  8, 0},
    {128, 128,  8,  8, 1},
};

extern "C" void kernel_launch(void* const* d_in, const int* in_sizes, int n_in,
                              void* d_out, int out_size, void* d_ws, size_t ws_size,
                              hipStream_t stream) {
  (void)in_sizes; (void)n_in; (void)out_size; (void)ws_size;

  const float* x = (const float*)d_in[0];
  const float* aw[8];
  const float* gma[8];
  const float* bta[8];
  const float* mu[8];
  const float* var[8];
  for (int i = 0; i < 8; ++i) {
    aw[i]  = (const float*)d_in[1 + i];
    gma[i] = (const float*)d_in[9 + i];
    bta[i] = (const float*)d_in[17 + i];
    mu[i]  = (const float*)d_in[25 + i];
    var[i] = (const float*)d_in[33 + i];
  }
  const float* fw = (const float*)d_in[41];
  const float* fb = (const float*)d_in[42];

  // Ping-pong activation buffers in workspace (max activation = 2 MB).
  float* bufA = (float*)d_ws;
  float* bufB = (float*)((char*)d_ws + (4u << 20));

  const float* cur = x;
  float* nxt = bufA;
  for (int i = 0; i < 8; ++i) {
    const LayerCfg& c = kCfg[i];
    const int Hp = c.H + 2, Wp = c.W + 2;
    size_t shmem = (size_t)(c.Cin * Hp * Wp + c.Cin * 9 +
                            (c.pool ? c.H * c.W : 0)) * sizeof(float);
    dim3 grid(16 * c.Cout);
    dim3 block(256);
    if (c.pool) {
      adder_bn_relu_kernel<true><<<grid, block, shmem, stream>>>(
          cur, aw[i], gma[i], bta[i], mu[i], var[i], nxt,
          c.Cin, c.Cout, c.H, c.W);
    } else {
      adder_bn_relu_kernel<false><<<grid, block, shmem, stream>>>(
          cur, aw[i], gma[i], bta[i], mu[i], var[i], nxt,
          c.Cin, c.Cout, c.H, c.W);
    }
    cur = nxt;
    nxt = (nxt == bufA) ? bufB : bufA;
  }

  // cur now holds [16,128,4,4] == flattened [16,2048] (NCHW reshape order).
  // Dynamic LDS: A tile + padded B tile = 2 * 16*2048 * 4B = 256 KB.
  fc_wmma_kernel<<<dim3(1), dim3(256), 2u * 16 * 2048 * sizeof(float), stream>>>(
      cur, fw, fb, (float*)d_out);
}